// STEM_18348100289194
// MI455X (gfx1250) — compile-verified
//
#include <hip/hip_runtime.h>
#include <hip/hip_bf16.h>

typedef __attribute__((ext_vector_type(2))) float v2f;
typedef __attribute__((ext_vector_type(8))) float v8f;
typedef __attribute__((ext_vector_type(4))) unsigned int v4u;
typedef __attribute__((ext_vector_type(4))) int v4i;
typedef __attribute__((ext_vector_type(8))) int v8i;

#define THREADS 256
#define BM      128                 // 8 waves x 16 rows
#define BN      112                 // 7 x 16 cols
#define KK      196                 // 14*14
#define NTILES  7

#define A_FLOATS (BM * KK)          // 25088 floats (patched [128][196])
#define W_FLOATS (BN * KK)          // 21952 floats ([112][196])

// TDM: issue one tensor_load_to_lds. Toolchains differ in builtin arity.
__device__ __forceinline__ void tdm_load(v4u g0, v8i g1, v4i g2, v4i g3) {
#if defined(__clang_major__) && (__clang_major__ >= 23)
    v8i g4 = {};
    __builtin_amdgcn_tensor_load_to_lds(g0, g1, g2, g3, g4, 0);
#else
    __builtin_amdgcn_tensor_load_to_lds(g0, g1, g2, g3, 0);
#endif
}

// out[m, n] = sum_k A[m,k] * W[n,k] + b[n]
// A[m,k] = x[m>>2, ((m&3)>>1)*14 + k/14, ((m&3)&1)*14 + k%14]
__global__ __launch_bounds__(THREADS)
void patch_linear_wmma_f32(const float* __restrict__ x,
                           const float* __restrict__ Wlin,
                           const float* __restrict__ blin,
                           float* __restrict__ out)
{
    extern __shared__ float smem[];
    float* sA = smem;               // patched A tile [128][196]
    float* sW = smem + A_FLOATS;    // W tile [112][196] (N x K)

    const int tid  = threadIdx.x;
    const int lane = tid & 31;
    const int wave = tid >> 5;      // 0..7

    const int n0      = blockIdx.x * BN;
    const int imgBase = blockIdx.y * 32;             // 32 whole images per tile
    const int m_base  = blockIdx.y * BM + wave * 16;

    // ---- TDM fill. The unfold/patch reorg is a single 5D tile move:
    // LDS-linear (fastest->slowest) = j(14,s=1), i(14,s=28), px(2,s=14),
    // py(2,s=392), img(32,s=784)  ==>  sA[m*196+k] with m=img*4+py*2+px,
    // k=i*14+j. Descriptor per ISA §8.3/8.4 (data_size=4B, type=2, count=1).
    if (tid == 0) {
        const unsigned long long ga =
            (unsigned long long)(uintptr_t)(x + (size_t)imgBase * 784);
        const unsigned int ldsA = (unsigned int)(uintptr_t)sA;
        v4u g0 = { 1u, ldsA, (unsigned int)ga,
                   (unsigned int)((ga >> 32) & 0x01FFFFFFu) | 0x80000000u };
        v8i g1 = { (int)0x00020000u,        // data_size = 2 (4B)
                   (int)(14u << 16),        // tensor_dim0 = 14
                   (int)(14u << 16),        // tensor_dim1.lo = 14
                   (int)(14u << 16),        // tile_dim0 = 14 (j)
                   (int)(14u | (2u << 16)), // tile_dim1 = 14 (i), tile_dim2 = 2 (px)
                   28,                      // tensor_dim0_stride = 28 (i stride)
                   (int)(14u << 16),        // tensor_dim1_stride.lo = 14 (px stride)
                   0 };
        v4i g2 = { 2,                       // tensor_dim2 = 2
                   2,                       // tensor_dim3 = 2
                   392,                     // tensor_dim2_stride = 392 (py stride)
                   (int)(2u << 16) };       // tile_dim3 = 2 (py)
        v4i g3 = { 784,                     // tensor_dim3_stride = 784 (img stride)
                   (int)(32u << 16),        // tensor_dim4 = 32
                   (int)(32u << 16),        // tile_dim4 = 32 (img)
                   0 };
        tdm_load(g0, g1, g2, g3);
        __builtin_amdgcn_s_wait_tensorcnt(0);
    } else if (tid == 32) {
        // W tile: plain 2D 196x112 contiguous copy.
        const unsigned long long ga =
            (unsigned long long)(uintptr_t)(Wlin + (size_t)n0 * KK);
        const unsigned int ldsW = (unsigned int)(uintptr_t)sW;
        v4u g0 = { 1u, ldsW, (unsigned int)ga,
                   (unsigned int)((ga >> 32) & 0x01FFFFFFu) | 0x80000000u };
        v8i g1 = { (int)0x00020000u,        // data_size = 4B
                   (int)(196u << 16),       // tensor_dim0 = 196
                   (int)(112u << 16),       // tensor_dim1 = 112
                   (int)(196u << 16),       // tile_dim0 = 196
                   112,                     // tile_dim1 = 112, tile_dim2 = 0
                   196,                     // tensor_dim0_stride = 196
                   0, 0 };
        v4i g2 = {};
        v4i g3 = {};
        tdm_load(g0, g1, g2, g3);
        __builtin_amdgcn_s_wait_tensorcnt(0);
    }
    __syncthreads();

    // Fragment base pointers. A layout (32-bit A 16x4): lanes 0-15 = M rows,
    // K pair (k0,k0+1); lanes 16-31 same M, K pair (k0+2,k0+3). B (4x16):
    // lanes 0-15 = N cols. Strides of 196 dwords are bank-conflict-free
    // (196 mod 64 = 4; 16 lanes x step 4 -> 16 distinct banks per half-wave).
    const int  khalf = (lane >> 4) * 2;
    const float* aPtr = sA + (wave * 16 + (lane & 15)) * KK + khalf;
    const float* bPtr = sW + (lane & 15) * KK + khalf;

    const v8f vzero = {};
    v8f acc[NTILES];
#pragma unroll
    for (int t = 0; t < NTILES; ++t) acc[t] = vzero;

    // K loop: 49 steps of 4. A address advances by a constant 16B/step ->
    // folds into ds_load immediate offsets (no per-step address VALU).
    for (int k0 = 0; k0 < KK; k0 += 4) {
        const v2f afrag = *(const v2f*)(aPtr + k0);
#pragma unroll
        for (int t = 0; t < NTILES; ++t) {
            const v2f bfrag = *(const v2f*)(bPtr + t * 16 * KK + k0);
            acc[t] = __builtin_amdgcn_wmma_f32_16x16x4_f32(
                false, afrag, false, bfrag, (short)0, acc[t], false, false);
        }
    }

    // D layout (16x16 f32): VGPR v -> row v (lanes 0-15) / row 8+v (16-31),
    // N = lane & 15.
    const int mrow = m_base + (lane >> 4) * 8;
#pragma unroll
    for (int t = 0; t < NTILES; ++t) {
        const int n = n0 + t * 16 + (lane & 15);
        const float bias = blin[n];
        float* o = out + (size_t)mrow * 784 + n;
#pragma unroll
        for (int v = 0; v < 8; ++v)
            o[(size_t)v * 784] = acc[t][v] + bias;
    }
}

extern "C" void kernel_launch(void* const* d_in, const int* in_sizes, int n_in,
                              void* d_out, int out_size, void* d_ws, size_t ws_size,
                              hipStream_t stream) {
    const float* x    = (const float*)d_in[0];   // [32768,1,28,28]
    const float* Wlin = (const float*)d_in[1];   // [784,196]
    const float* blin = (const float*)d_in[2];   // [784]
    float* out = (float*)d_out;                  // [32768,4,784]

    const size_t smem = (size_t)(A_FLOATS + W_FLOATS) * sizeof(float); // 188160 B
    (void)hipFuncSetAttribute(reinterpret_cast<const void*>(&patch_linear_wmma_f32),
                              hipFuncAttributeMaxDynamicSharedMemorySize, (int)smem);

    dim3 grid(784 / BN, (32768 * 4) / BM);       // (7, 1024)
    patch_linear_wmma_f32<<<grid, THREADS, smem, stream>>>(x, Wlin, blin, out);
}